// ShiftNetwork_35046933135690
// MI455X (gfx1250) — compile-verified
//
#include <hip/hip_runtime.h>
#include <hip/hip_bf16.h>
#include <math.h>

// ---------------------------------------------------------------------------
// ShiftNetwork on MI455X (gfx1250, wave32)
//
// Phase 1 (dominant, ~167.7MB streamed => ~7.2us @ 23.3TB/s):
//   masked column mean over 2047 rotated rows of X. 2-D grid: columns x
//   row-slices, partial sums in workspace, non-temporal loads (read-once).
// Phase 2: forward-fill + stride-10 sample -> xs[2048] (trivial).
// Phase 3: MLP. GEMM layers use V_WMMA_F32_16X16X4_F32 (fp32 WMMA) to keep
//   full precision; one wave per 16x16 tile, K-loop of 64 WMMAs.
// ---------------------------------------------------------------------------

#define SRD   10
#define NCOL  20480      // C
#define PN    2048       // P
#define PROWS 2047       // P-1 contributing rows
#define HID   256        // hidden size
#define NY    32         // row slices for partial reduction

typedef __attribute__((ext_vector_type(2))) float v2f;
typedef __attribute__((ext_vector_type(8))) float v8f;

// ---- Phase 1a: per-slice partial column sums/counts of rotated rows -------
__global__ void col_partial_kernel(const float* __restrict__ X,
                                   const int*   __restrict__ p,
                                   float* __restrict__ psum,
                                   int*   __restrict__ pcnt,
                                   int rowsPerSlice) {
    const int j  = blockIdx.x * blockDim.x + threadIdx.x;   // output column
    const int y  = blockIdx.y;
    int i0 = y * rowsPerSlice;
    int i1 = i0 + rowsPerSlice;
    if (i1 > PROWS) i1 = PROWS;

    float s = 0.0f;
    int   c = 0;
    for (int i = i0; i < i1; ++i) {
        const int pi  = p[i];                 // uniform -> scalar load
        int src = j - pi;                     // roll right by p[i]
        if (src < 0) src += NCOL;
        // X row i*SR, streamed once: non-temporal load
        const float v = __builtin_nontemporal_load(
            X + (size_t)i * (size_t)SRD * (size_t)NCOL + (size_t)src);
        if (!(v != v)) { s += v; ++c; }       // NaN-masked accumulate
    }
    psum[(size_t)y * NCOL + j] = s;
    pcnt[(size_t)y * NCOL + j] = c;
}

// ---- Phase 1b: combine partials -> masked column mean m[NCOL] -------------
__global__ void col_mean_kernel(const float* __restrict__ psum,
                                const int*   __restrict__ pcnt,
                                float* __restrict__ m) {
    const int j = blockIdx.x * blockDim.x + threadIdx.x;
    float s = 0.0f;
    int   c = 0;
    #pragma unroll 4
    for (int y = 0; y < NY; ++y) {
        s += psum[(size_t)y * NCOL + j];
        c += pcnt[(size_t)y * NCOL + j];
    }
    m[j] = (c > 0) ? (s / (float)c) : __builtin_nanf("");
}

// ---- Phase 2: forward-fill NaNs + sample every SR-th column ---------------
// filled[j] = m at greatest j' <= j with m[j'] non-NaN (else m[0]).
// Lazy backward scan: 1 load in the common (no-NaN) case.
__global__ void sample_ffill_kernel(const float* __restrict__ m,
                                    float* __restrict__ xs) {
    const int t = blockIdx.x * blockDim.x + threadIdx.x;   // 0..PN-1
    int j = t * SRD;
    float v = m[j];
    while ((v != v) && j > 0) { --j; v = m[j]; }
    xs[t] = v;
}

// ---- Phase 3a: layer 1 (outer product, no GEMM needed) --------------------
__global__ void layer1_kernel(const float* __restrict__ xs,
                              const float* __restrict__ w1,   // [HID,1]
                              const float* __restrict__ b1,
                              float* __restrict__ h1) {       // [PN,HID]
    const int t = blockIdx.x;        // row
    const int c = threadIdx.x;       // hidden unit
    float r = xs[t] * w1[c] + b1[c];
    h1[(size_t)t * HID + c] = r > 0.0f ? r : 0.0f;
}

// ---- Phase 3b: GEMM + bias + ReLU via V_WMMA_F32_16X16X4_F32 --------------
// out[r][n] = relu( sum_k Hin[r][k] * W[n][k] + bias[n] )   (Hin @ W^T)
// One wave per 16(M)x16(N) tile; K=256 in steps of 4.
// A frag (16x4): lane l<16 -> K={0,1}, l>=16 -> K={2,3}; M = l%16.
// B frag (4x16): mirrored, N = l%16.
// C/D: VGPR v holds row m0 + v + 8*(lane/16), col n0 + lane%16.
__global__ void gemm_relu_wmma_kernel(const float* __restrict__ Hin,  // [M,HID]
                                      const float* __restrict__ W,    // [HID,HID]
                                      const float* __restrict__ bias, // [HID]
                                      float* __restrict__ Hout,       // [M,HID]
                                      int M) {
    const int wave = (blockIdx.x * blockDim.x + threadIdx.x) >> 5;
    const int lane = threadIdx.x & 31;
    const int tilesN = HID / 16;                 // 16
    const int m0 = (wave / tilesN) * 16;
    const int n0 = (wave % tilesN) * 16;
    if (m0 >= M) return;                         // wave-uniform; EXEC stays full

    const int l16 = lane & 15;
    const int kh  = lane >> 4;                   // 0 or 1
    const float* __restrict__ arow = Hin + (size_t)(m0 + l16) * HID + 2 * kh;
    const float* __restrict__ brow = W   + (size_t)(n0 + l16) * HID + 2 * kh;

    v8f acc = {};
    #pragma unroll 8
    for (int k = 0; k < HID; k += 4) {
        v2f a, b;
        a.x = arow[k];  a.y = arow[k + 1];       // -> global_load_b64
        b.x = brow[k];  b.y = brow[k + 1];
        acc = __builtin_amdgcn_wmma_f32_16x16x4_f32(
            /*neg_a=*/false, a, /*neg_b=*/false, b,
            /*c_mod=*/(short)0, acc, /*reuse_a=*/false, /*reuse_b=*/false);
    }

    const float bb = bias[n0 + l16];
    const int   colIdx = n0 + l16;
    const int   rbase  = m0 + 8 * kh;
    #pragma unroll
    for (int v = 0; v < 8; ++v) {
        float r = acc[v] + bb;
        r = r > 0.0f ? r : 0.0f;
        Hout[(size_t)(rbase + v) * HID + colIdx] = r;
    }
}

// ---- Phase 3c: final layer: per-row dot with w4 + b4 ----------------------
__global__ void layer4_kernel(const float* __restrict__ h3,  // [PN,HID]
                              const float* __restrict__ w4,  // [1,HID]
                              const float* __restrict__ b4,  // [1]
                              float* __restrict__ out) {     // [PN]
    const int wave = (blockIdx.x * blockDim.x + threadIdx.x) >> 5;
    const int lane = threadIdx.x & 31;
    const float* row = h3 + (size_t)wave * HID;
    float s = 0.0f;
    #pragma unroll
    for (int k = lane; k < HID; k += 32) s += row[k] * w4[k];
    #pragma unroll
    for (int off = 16; off > 0; off >>= 1) s += __shfl_down(s, off, 32);
    if (lane == 0) out[wave] = s + b4[0];
}

// ---------------------------------------------------------------------------
extern "C" void kernel_launch(void* const* d_in, const int* in_sizes, int n_in,
                              void* d_out, int out_size, void* d_ws, size_t ws_size,
                              hipStream_t stream) {
    const float* X  = (const float*)d_in[0];
    const int*   p  = (const int*)  d_in[1];
    const float* w1 = (const float*)d_in[2];
    const float* b1 = (const float*)d_in[3];
    const float* w2 = (const float*)d_in[4];
    const float* b2 = (const float*)d_in[5];
    const float* w3 = (const float*)d_in[6];
    const float* b3 = (const float*)d_in[7];
    const float* w4 = (const float*)d_in[8];
    const float* b4 = (const float*)d_in[9];
    float* out = (float*)d_out;

    // Workspace layout (floats). Regions 0/1 are reused for the MLP
    // activations after the reduction phase (524288 <= 655360, no aliasing
    // between any layer's input and output).
    float* ws    = (float*)d_ws;
    float* psum  = ws;                                  // [NY*NCOL]
    int*   pcnt  = (int*)(ws + (size_t)NY * NCOL);      // [NY*NCOL]
    float* m     = ws + 2 * (size_t)NY * NCOL;          // [NCOL]
    float* xs    = m + NCOL;                            // [PN]
    float* h1    = psum;                                // [PN*HID] reuse
    float* h2    = (float*)pcnt;                        // [PN*HID] reuse
    float* h3    = psum;                                // reuse again (h1 dead)

    const int rowsPerSlice = (PROWS + NY - 1) / NY;     // 64

    // Phase 1: masked column mean (the bandwidth-bound part)
    col_partial_kernel<<<dim3(NCOL / 256, NY), 256, 0, stream>>>(
        X, p, psum, pcnt, rowsPerSlice);
    col_mean_kernel<<<NCOL / 256, 256, 0, stream>>>(psum, pcnt, m);

    // Phase 2: forward-fill + sample
    sample_ffill_kernel<<<PN / 256, 256, 0, stream>>>(m, xs);

    // Phase 3: MLP
    layer1_kernel<<<PN, HID, 0, stream>>>(xs, w1, b1, h1);

    const int gemmWaves   = (PN / 16) * (HID / 16);     // 2048 waves
    const int gemmBlocks  = gemmWaves * 32 / 256;       // 256 blocks
    gemm_relu_wmma_kernel<<<gemmBlocks, 256, 0, stream>>>(h1, w2, b2, h2, PN);
    gemm_relu_wmma_kernel<<<gemmBlocks, 256, 0, stream>>>(h2, w3, b3, h3, PN);

    layer4_kernel<<<PN * 32 / 256, 256, 0, stream>>>(h3, w4, b4, out);

    (void)in_sizes; (void)n_in; (void)out_size; (void)ws_size;
}